// SeriesHMMTinyMoARNN_14104672600841
// MI455X (gfx1250) — compile-verified
//
#include <hip/hip_runtime.h>
#include <hip/hip_bf16.h>
#include <math.h>

// Problem constants (match reference)
#define BB 256
#define TT 4096
#define DD 3
#define KK 2
#define HH 2
#define AA 5
#define EHH 8

typedef __attribute__((ext_vector_type(2))) float v2f;
typedef __attribute__((ext_vector_type(8))) float v8f;

__device__ __forceinline__ float lse2(float u, float v) {
    float m = fmaxf(u, v);
    return m + logf(expf(u - m) + expf(v - m));
}
__device__ __forceinline__ float sigmoidf(float x) {
    return 1.0f / (1.0f + expf(-x));
}

// ---------------------------------------------------------------------------
// Kernel 1: fused parallel pre-pass. One wave handles 16 flattened (b,t) rows
// with a single V_WMMA_F32_16X16X4_F32 whose B matrix packs TWO weight sets:
//   cols 0..7  : fc1_w^T  (emitter hidden layer,   x @ fc1_w.T)
//   cols 8..13 : w_ih^T   (GRU input-gate preacts, x @ w_ih.T)
//   cols 14,15 : zero
// A (16x4 f32): 16 rows of x, K = D(=3) padded to 4.
//   ISA 7.12.2: lane l (0..15) holds A[l,0],A[l,1]; lane l+16 holds A[l,2],pad.
// D (16x16 f32): VGPR r: lanes 0-15 -> (M=r, N=lane), lanes 16-31 -> (M=r+8).
// Emitter second layer (8 -> 2) via shfl_xor column reductions per half-wave
// (gi columns carry zero weights there, so they never pollute the sums).
// Writes eT [T,B,K] and giT [T,B,6] (gi includes b_ih) for the scan phase.
// ---------------------------------------------------------------------------
__global__ void emit_kernel(const float* __restrict__ x,
                            const float* __restrict__ fc1_w, const float* __restrict__ fc1_b,
                            const float* __restrict__ fc2_w, const float* __restrict__ fc2_b,
                            const float* __restrict__ w_ih,  const float* __restrict__ b_ih,
                            float* __restrict__ eT, float* __restrict__ giT) {
    const int lane = threadIdx.x & 31;
    const int wave = threadIdx.x >> 5;
    const long tile = (long)blockIdx.x * (blockDim.x >> 5) + wave;
    const long row0 = tile * 16;            // 16 consecutive flattened (b,t) rows
    const int half  = lane >> 4;            // 0: K=0,1   1: K=2,3
    const int l     = lane & 15;

    // ---- A matrix: rows of x, K padded 3 -> 4 ----
    const long rowA = row0 + l;
    v2f a;
    if (half == 0) { a[0] = x[rowA * 3 + 0]; a[1] = x[rowA * 3 + 1]; }
    else           { a[0] = x[rowA * 3 + 2]; a[1] = 0.0f; }

    // ---- B matrix: [fc1_w^T | w_ih^T | 0] packed into 4 x 16 ----
    v2f bm; bm[0] = 0.0f; bm[1] = 0.0f;
    if (l < EHH) {                               // cols 0..7 : fc1_w^T
        if (half == 0) { bm[0] = fc1_w[l * 3 + 0]; bm[1] = fc1_w[l * 3 + 1]; }
        else           { bm[0] = fc1_w[l * 3 + 2]; }
    } else if (l < EHH + 6) {                    // cols 8..13 : w_ih^T
        const int j = l - EHH;
        if (half == 0) { bm[0] = w_ih[j * 3 + 0]; bm[1] = w_ih[j * 3 + 1]; }
        else           { bm[0] = w_ih[j * 3 + 2]; }
    }

    v8f acc = {};
    acc = __builtin_amdgcn_wmma_f32_16x16x4_f32(
        /*neg_a=*/false, a, /*neg_b=*/false, bm,
        /*c_mod=*/(short)0, acc, /*reuse_a=*/false, /*reuse_b=*/false);

    // ---- GRU input-gate preacts: cols 8..13 -> giT [T,B,6] (bias folded) ----
    if (l >= EHH && l < EHH + 6) {
        const int j = l - EHH;
        const float bj = b_ih[j];
#pragma unroll
        for (int r = 0; r < 8; ++r) {
            long row = row0 + r + half * 8;
            long bb_ = row >> 12;           // row / T   (T = 4096)
            long tt_ = row & (TT - 1);      // row % T
            giT[(tt_ * BB + bb_) * 6 + j] = acc[r] + bj;
        }
    }

    // ---- emitter hidden + 2nd layer partials (column n = l, cols >= 8 zeroed) ----
    const float fb = (l < EHH) ? fc1_b[l]       : 0.0f;
    const float w0 = (l < EHH) ? fc2_w[l]       : 0.0f;   // fc2_w[0*EH + n]
    const float w1 = (l < EHH) ? fc2_w[EHH + l] : 0.0f;   // fc2_w[1*EH + n]
    float s0[8], s1[8];
#pragma unroll
    for (int r = 0; r < 8; ++r) {
        float tv = (l < EHH) ? tanhf(acc[r] + fb) : 0.0f;
        s0[r] = tv * w0;
        s1[r] = tv * w1;
    }
    // reduce over the 16 columns inside each 16-lane half
#pragma unroll
    for (int m = 1; m < 16; m <<= 1) {
#pragma unroll
        for (int r = 0; r < 8; ++r) {
            s0[r] += __shfl_xor(s0[r], m, 32);
            s1[r] += __shfl_xor(s1[r], m, 32);
        }
    }
    // distributed log-softmax: lane l (<8) of each half handles row M = l + 8*half
    if (l < 8) {
        float y0 = s0[0], y1 = s1[0];
#pragma unroll
        for (int r = 1; r < 8; ++r) { if (l == r) { y0 = s0[r]; y1 = s1[r]; } }
        y0 += fc2_b[0];
        y1 += fc2_b[1];
        float mx = fmaxf(y0, y1);
        float z  = mx + logf(expf(y0 - mx) + expf(y1 - mx));
        long row = row0 + l + half * 8;
        long bb_ = row >> 12;
        long tt_ = row & (TT - 1);
        long o = (tt_ * BB + bb_) * 2;
        eT[o + 0] = y0 - z;
        eT[o + 1] = y1 - z;
    }
}

// ---------------------------------------------------------------------------
// Kernel 2: the three sequential scans, run CONCURRENTLY on different WGPs.
//   blockIdx.x == 0 : GRU over T (one batch chain per lane; gi precomputed)
//   blockIdx.x == 1 : HMM forward  (alpha)
//   blockIdx.x == 2 : HMM backward (beta)
// All state arrays are [T,B,*] so consecutive lanes (b) hit consecutive memory.
// ---------------------------------------------------------------------------
__global__ void scans_kernel(const float* __restrict__ giT,
                             const float* __restrict__ w_hh, const float* __restrict__ b_hh,
                             const float* __restrict__ log_pi, const float* __restrict__ log_A,
                             const float* __restrict__ eT,
                             float* __restrict__ hT, float* __restrict__ alpha,
                             float* __restrict__ beta) {
    const int b = threadIdx.x;
    if (b >= BB) return;

    if (blockIdx.x == 0) {
        // ---------------- GRU (h-dependent part only) ----------------
        float whh[6][2], bh[6];
#pragma unroll
        for (int j = 0; j < 6; ++j) {
#pragma unroll
            for (int d = 0; d < 2; ++d) whh[j][d] = w_hh[j * 2 + d];
            bh[j] = b_hh[j];
        }
        float h0 = 0.0f, h1 = 0.0f;
        const float* gp = giT + (long)b * 6;   // t = 0 row
        for (int t = 0; t < TT; ++t) {
            float gi0 = gp[0], gi1 = gp[1], gi2 = gp[2];
            float gi3 = gp[3], gi4 = gp[4], gi5 = gp[5];
            __builtin_prefetch(gp + 6 * BB, 0, 0);   // global_prefetch: next timestep
            gp += 6 * BB;
            float gh0 = whh[0][0]*h0 + whh[0][1]*h1 + bh[0];
            float gh1 = whh[1][0]*h0 + whh[1][1]*h1 + bh[1];
            float gh2 = whh[2][0]*h0 + whh[2][1]*h1 + bh[2];
            float gh3 = whh[3][0]*h0 + whh[3][1]*h1 + bh[3];
            float gh4 = whh[4][0]*h0 + whh[4][1]*h1 + bh[4];
            float gh5 = whh[5][0]*h0 + whh[5][1]*h1 + bh[5];
            float r0 = sigmoidf(gi0 + gh0);
            float r1 = sigmoidf(gi1 + gh1);
            float z0 = sigmoidf(gi2 + gh2);
            float z1 = sigmoidf(gi3 + gh3);
            float n0 = tanhf(gi4 + r0 * gh4);
            float n1 = tanhf(gi5 + r1 * gh5);
            h0 = (1.0f - z0) * n0 + z0 * h0;
            h1 = (1.0f - z1) * n1 + z1 * h1;
            long o = ((long)t * BB + b) * 2;
            hT[o + 0] = h0;
            hT[o + 1] = h1;
        }
    } else {
        // shared transition normalization (K=2)
        float A00 = log_A[0], A01 = log_A[1], A10 = log_A[2], A11 = log_A[3];
        float z0 = lse2(A00, A01); A00 -= z0; A01 -= z0;
        float z1 = lse2(A10, A11); A10 -= z1; A11 -= z1;

        if (blockIdx.x == 1) {
            // ---------------- HMM forward (alpha) ----------------
            float p0 = log_pi[0], p1 = log_pi[1];
            float zp = lse2(p0, p1);
            p0 -= zp; p1 -= zp;
            float a0 = p0 + eT[b * 2 + 0];
            float a1 = p1 + eT[b * 2 + 1];
            alpha[b * 2 + 0] = a0;
            alpha[b * 2 + 1] = a1;
            for (int t = 1; t < TT; ++t) {
                const float* ep = eT + ((long)t * BB + b) * 2;
                __builtin_prefetch(ep + 2 * BB, 0, 0);
                float e0 = ep[0], e1 = ep[1];
                float n0 = lse2(a0 + A00, a1 + A10) + e0;
                float n1 = lse2(a0 + A01, a1 + A11) + e1;
                a0 = n0; a1 = n1;
                long o = ((long)t * BB + b) * 2;
                alpha[o + 0] = a0;
                alpha[o + 1] = a1;
            }
        } else {
            // ---------------- HMM backward (beta) ----------------
            float bb0 = 0.0f, bb1 = 0.0f;
            long oN = ((long)(TT - 1) * BB + b) * 2;
            beta[oN + 0] = 0.0f;
            beta[oN + 1] = 0.0f;
            for (int t = TT - 2; t >= 0; --t) {
                const float* ep = eT + ((long)(t + 1) * BB + b) * 2;
                __builtin_prefetch(ep - 2 * BB, 0, 0);
                float e0 = ep[0] + bb0;   // b_next + e[t+1]
                float e1 = ep[1] + bb1;
                float n0 = lse2(e0 + A00, e1 + A01);
                float n1 = lse2(e0 + A10, e1 + A11);
                bb0 = n0; bb1 = n1;
                long o = ((long)t * BB + b) * 2;
                beta[o + 0] = bb0;
                beta[o + 1] = bb1;
            }
        }
    }
}

// ---------------------------------------------------------------------------
// Kernel 3: fused posterior + mixture-of-agents gating + outputs.
// One thread per flattened (b,t) row (row = b*T + t so Q_seq reads and all
// output writes are fully coalesced; [T,B,*] workspace reads hit L2).
// out layout: [pi_log (B*T*2) | g (B*T*5) | log_gamma (B*T*2)]
// ---------------------------------------------------------------------------
__global__ void gate_kernel(const float* __restrict__ Q_seq, const float* __restrict__ Wg,
                            const float* __restrict__ by,
                            const float* __restrict__ hT, const float* __restrict__ alpha,
                            const float* __restrict__ beta, float* __restrict__ out) {
    const long row = (long)blockIdx.x * blockDim.x + threadIdx.x;
    if (row >= (long)BB * TT) return;
    const long b = row >> 12;        // row / T
    const long t = row & (TT - 1);   // row % T
    const long w = (t * BB + b) * 2;

    // posterior log_gamma / gamma
    float lg0 = alpha[w + 0] + beta[w + 0];
    float lg1 = alpha[w + 1] + beta[w + 1];
    float z = lse2(lg0, lg1);
    lg0 -= z; lg1 -= z;
    float g0 = expf(lg0), g1 = expf(lg1);

    const float h0 = hT[w + 0], h1 = hT[w + 1];

    // gating: g[a] = sum_k gamma_k * softmax_a(h . Wg[k,:,a])
    float ga[AA] = {0.0f, 0.0f, 0.0f, 0.0f, 0.0f};
#pragma unroll
    for (int k = 0; k < KK; ++k) {
        const float gam = (k == 0) ? g0 : g1;
        const float* Wk = Wg + k * (HH * AA);
        float wh[AA];
        float mx = -INFINITY;
#pragma unroll
        for (int a = 0; a < AA; ++a) {
            wh[a] = h0 * Wk[a] + h1 * Wk[AA + a];
            mx = fmaxf(mx, wh[a]);
        }
        float s = 0.0f;
#pragma unroll
        for (int a = 0; a < AA; ++a) { wh[a] = expf(wh[a] - mx); s += wh[a]; }
        const float inv = gam / s;
#pragma unroll
        for (int a = 0; a < AA; ++a) ga[a] += wh[a] * inv;
    }

    // V = g @ Q_seq[b,t] + gamma @ by ; pi_log = log_softmax(V)
    const float* qp = Q_seq + row * (AA * 2);
    float V0 = 0.0f, V1 = 0.0f;
#pragma unroll
    for (int a = 0; a < AA; ++a) {
        V0 += ga[a] * qp[a * 2 + 0];
        V1 += ga[a] * qp[a * 2 + 1];
    }
    V0 += g0 * by[0] + g1 * by[2];
    V1 += g0 * by[1] + g1 * by[3];
    float mv = fmaxf(V0, V1);
    float lz = mv + logf(expf(V0 - mv) + expf(V1 - mv));

    // outputs
    out[row * 2 + 0] = V0 - lz;
    out[row * 2 + 1] = V1 - lz;
    float* gout = out + (long)BB * TT * 2;
#pragma unroll
    for (int a = 0; a < AA; ++a) gout[row * AA + a] = ga[a];
    float* lgout = out + (long)BB * TT * 7;
    lgout[row * 2 + 0] = lg0;
    lgout[row * 2 + 1] = lg1;
}

// ---------------------------------------------------------------------------
extern "C" void kernel_launch(void* const* d_in, const int* in_sizes, int n_in,
                              void* d_out, int out_size, void* d_ws, size_t ws_size,
                              hipStream_t stream) {
    const float* x      = (const float*)d_in[0];
    const float* Q_seq  = (const float*)d_in[1];
    const float* log_pi = (const float*)d_in[2];
    const float* log_A  = (const float*)d_in[3];
    const float* fc1_w  = (const float*)d_in[4];
    const float* fc1_b  = (const float*)d_in[5];
    const float* fc2_w  = (const float*)d_in[6];
    const float* fc2_b  = (const float*)d_in[7];
    const float* w_ih   = (const float*)d_in[8];
    const float* w_hh   = (const float*)d_in[9];
    const float* b_ih   = (const float*)d_in[10];
    const float* b_hh   = (const float*)d_in[11];
    const float* Wg     = (const float*)d_in[12];
    const float* by     = (const float*)d_in[13];
    float* out = (float*)d_out;

    // workspace: hT | eT | alpha | beta (each [T,B,2]) | giT ([T,B,6])  = 56 MB
    float* ws    = (float*)d_ws;
    const size_t slab = (size_t)TT * BB * 2;
    float* hT    = ws;
    float* eT    = ws + slab;
    float* alpha = ws + 2 * slab;
    float* beta  = ws + 3 * slab;
    float* giT   = ws + 4 * slab;

    // 1) fused WMMA pre-pass: emitter log-probs + GRU input gates
    const int tiles = (BB * TT) / 16;           // 65536 tiles, 8 waves per block
    emit_kernel<<<dim3(tiles / 8), dim3(256), 0, stream>>>(
        x, fc1_w, fc1_b, fc2_w, fc2_b, w_ih, b_ih, eT, giT);

    // 2) the three sequential scans, concurrently (3 blocks on 3 WGPs)
    scans_kernel<<<dim3(3), dim3(256), 0, stream>>>(
        giT, w_hh, b_hh, log_pi, log_A, eT, hT, alpha, beta);

    // 3) fused gating + outputs
    const long rows = (long)BB * TT;
    gate_kernel<<<dim3((unsigned)((rows + 255) / 256)), dim3(256), 0, stream>>>(
        Q_seq, Wg, by, hT, alpha, beta, out);
}